// DENS_Encoder_35003983462543
// MI455X (gfx1250) — compile-verified
//
#include <hip/hip_runtime.h>

// ---- problem constants (from reference) ----
#define U_N       100000
#define I_N       50000
#define N_TOT     150000
#define D_DIM     128
#define NNZ_N     3200000
#define OUT_STRIDE 512      // (n_layers+1)*D = 4*128 floats per node in d_out
#define N_LAYERS  3
#define CAP       2048      // LDS staging capacity (nnz) per 8-row block segment

typedef unsigned int       u32;
typedef unsigned long long u64;
typedef u32 uint32x4_t __attribute__((ext_vector_type(4)));
typedef int int32x4_t  __attribute__((ext_vector_type(4)));
typedef int int32x8_t  __attribute__((ext_vector_type(8)));

// ---------------- TDM: 1-D tile of 4-byte elements, global -> LDS -------------
// D# group0: count=1, lds_addr, global_addr[56:0], type=2
// D# group1: data_size=2(4B), tensor_dim0=n, tensor_dim1=1, tile_dim0=n,
//            tile_dim1=1, tensor_dim0_stride=n, workgroup_mask=0 (no cluster)
__device__ __forceinline__ void tdm_load_1d_b32(u32 lds_off, const void* gptr, u32 n) {
#if defined(__HIP_DEVICE_COMPILE__)
  u64 ga = (u64)(size_t)gptr;
  uint32x4_t g0;
  g0.x = 1u;                                            // count=1, user mode
  g0.y = lds_off;                                       // lds_addr (bytes)
  g0.z = (u32)ga;                                       // global_addr[31:0]
  g0.w = ((u32)(ga >> 32) & 0x01FFFFFFu) | (2u << 30);  // addr[56:32] | type=2
  int32x8_t g1;
  g1[0] = (int)(2u << 16);                              // data_size = 4 bytes
  g1[1] = (int)((n & 0xFFFFu) << 16);                   // tensor_dim0[15:0]
  g1[2] = (int)(((n >> 16) & 0xFFFFu) | (1u << 16));    // tensor_dim0[31:16], tensor_dim1=1
  g1[3] = (int)((n & 0xFFFFu) << 16);                   // tile_dim0 = n (n <= CAP < 65536)
  g1[4] = 1;                                            // tile_dim1 = 1, tile_dim2 = 0
  g1[5] = (int)n;                                       // tensor_dim0_stride[31:0]
  g1[6] = 0;
  g1[7] = 0;
  int32x4_t z4 = {0, 0, 0, 0};
#if __clang_major__ >= 23
  int32x8_t z8 = {0, 0, 0, 0, 0, 0, 0, 0};
  __builtin_amdgcn_tensor_load_to_lds(g0, g1, z4, z4, z8, 0);
#else
  __builtin_amdgcn_tensor_load_to_lds(g0, g1, z4, z4, 0);
#endif
#endif
}

__device__ __forceinline__ void wait_tensor0() {
#if defined(__HIP_DEVICE_COMPILE__)
  __builtin_amdgcn_s_wait_tensorcnt(0);
#endif
}

// ---------------- layer 0: concat(user_emb, item_emb) into d_out --------------
__global__ void k_layer0(const float* __restrict__ ue, const float* __restrict__ ie,
                         float* __restrict__ out) {
  int tid  = blockIdx.x * blockDim.x + threadIdx.x;   // N*32 threads
  int node = tid >> 5;
  int q    = (tid & 31) << 2;
  if (node >= N_TOT) return;
  const float* src = (node < U_N) ? (ue + (size_t)node * D_DIM + q)
                                  : (ie + (size_t)(node - U_N) * D_DIM + q);
  float4 v = *(const float4*)src;
  *(float4*)(out + (size_t)node * OUT_STRIDE + q) = v;
}

// ---------------- CSR construction --------------------------------------------
__global__ void k_zero(int* __restrict__ p, int n) {
  int i = blockIdx.x * blockDim.x + threadIdx.x;
  if (i < n) p[i] = 0;
}

__global__ void k_hist(const int* __restrict__ rows, int* __restrict__ cnt) {
  int e = blockIdx.x * blockDim.x + threadIdx.x;
  if (e < NNZ_N) atomicAdd(&cnt[rows[e]], 1);
}

// single-workgroup chunked exclusive scan: ptr[0..n-1] counts -> offsets, ptr[n]=total
__global__ void __launch_bounds__(1024) k_scan(int* __restrict__ ptr,
                                               int* __restrict__ fill, int n) {
  __shared__ int sh[1024];
  __shared__ int run;
  int t = threadIdx.x;
  if (t == 0) run = 0;
  __syncthreads();
  for (int base = 0; base < n; base += 1024) {
    int i = base + t;
    int c = (i < n) ? ptr[i] : 0;
    sh[t] = c;
    __syncthreads();
    for (int off = 1; off < 1024; off <<= 1) {   // Hillis-Steele inclusive scan
      int v = (t >= off) ? sh[t - off] : 0;
      __syncthreads();
      sh[t] += v;
      __syncthreads();
    }
    int excl = sh[t] - c + run;
    if (i < n) { ptr[i] = excl; fill[i] = excl; }
    int tot = sh[1023];
    __syncthreads();
    if (t == 0) run += tot;
    __syncthreads();
  }
  if (t == 0) ptr[n] = run;
}

__global__ void k_scatter(const float* __restrict__ vals, const int* __restrict__ rows,
                          const int* __restrict__ cols, int* __restrict__ fill,
                          int* __restrict__ ccol, float* __restrict__ cval) {
  int e = blockIdx.x * blockDim.x + threadIdx.x;
  if (e >= NNZ_N) return;
  int r = rows[e];
  int p = atomicAdd(&fill[r], 1);
  ccol[p] = cols[e];
  cval[p] = vals[e];
}

// ---------------- CSR SpMM: one row per wave, TDM-staged CSR segment -----------
__global__ void __launch_bounds__(256) k_spmm(const int* __restrict__ row_ptr,
                                              const int* __restrict__ ccol,
                                              const float* __restrict__ cval,
                                              float* __restrict__ out, int layer) {
  __shared__ int   s_col[CAP];
  __shared__ float s_val[CAP];

  const int r0   = blockIdx.x * 8;          // 8 waves -> 8 consecutive rows
  const int wave = threadIdx.x >> 5;
  const int lane = threadIdx.x & 31;
  const int q    = lane << 2;               // lane owns floats [q, q+4)

  const int seg_s   = row_ptr[r0];          // uniform (r0 uniform) -> SGPR
  const int seg_e   = row_ptr[r0 + 8];
  const int seg_len = seg_e - seg_s;
  const bool lds_path = (seg_len <= CAP);

  if (lds_path && seg_len > 0 && threadIdx.x < 32) {
    tdm_load_1d_b32((u32)(size_t)&s_col[0], ccol + seg_s, (u32)seg_len);
    tdm_load_1d_b32((u32)(size_t)&s_val[0], cval + seg_s, (u32)seg_len);
    wait_tensor0();
  }
  __syncthreads();

  const int row   = r0 + wave;
  const int start = row_ptr[row];
  const int end   = row_ptr[row + 1];
  const float* __restrict__ xb = out + (size_t)(layer - 1) * D_DIM + q;

  float4 acc = {0.f, 0.f, 0.f, 0.f};
  if (lds_path) {
    const int lo = start - seg_s, hi = end - seg_s;
    #pragma unroll 4
    for (int j = lo; j < hi; ++j) {
      int   c = s_col[j];                   // LDS broadcast across the wave
      float v = s_val[j];
      float4 x = *(const float4*)(xb + (size_t)c * OUT_STRIDE);  // L2-resident gather
      acc.x = fmaf(v, x.x, acc.x);
      acc.y = fmaf(v, x.y, acc.y);
      acc.z = fmaf(v, x.z, acc.z);
      acc.w = fmaf(v, x.w, acc.w);
    }
  } else {                                  // (astronomically rare) big segment
    for (int j = start; j < end; ++j) {
      int   c = ccol[j];
      float v = cval[j];
      float4 x = *(const float4*)(xb + (size_t)c * OUT_STRIDE);
      acc.x = fmaf(v, x.x, acc.x);
      acc.y = fmaf(v, x.y, acc.y);
      acc.z = fmaf(v, x.z, acc.z);
      acc.w = fmaf(v, x.w, acc.w);
    }
  }
  *(float4*)(out + (size_t)row * OUT_STRIDE + (size_t)layer * D_DIM + q) = acc;
}

// ---------------- fallback path (no workspace): atomic COO SpMM ----------------
__global__ void k_zero_layer(float* __restrict__ out, int layer) {
  int tid  = blockIdx.x * blockDim.x + threadIdx.x;
  int node = tid >> 5;
  int q    = (tid & 31) << 2;
  if (node >= N_TOT) return;
  float4 z = {0.f, 0.f, 0.f, 0.f};
  *(float4*)(out + (size_t)node * OUT_STRIDE + (size_t)layer * D_DIM + q) = z;
}

__global__ void k_spmm_atomic(const float* __restrict__ vals, const int* __restrict__ rows,
                              const int* __restrict__ cols, float* __restrict__ out,
                              int layer) {
  int tid = blockIdx.x * blockDim.x + threadIdx.x;    // NNZ*32 threads
  int e   = tid >> 5;
  int q   = (tid & 31) << 2;
  if (e >= NNZ_N) return;
  int   r = rows[e];
  int   c = cols[e];
  float v = vals[e];
  const float* x = out + (size_t)(layer - 1) * D_DIM + (size_t)c * OUT_STRIDE + q;
  float4 xr = *(const float4*)x;
  float* o = out + (size_t)layer * D_DIM + (size_t)r * OUT_STRIDE + q;
  atomicAdd(o + 0, v * xr.x);
  atomicAdd(o + 1, v * xr.y);
  atomicAdd(o + 2, v * xr.z);
  atomicAdd(o + 3, v * xr.w);
}

// ---------------- host launch ---------------------------------------------------
extern "C" void kernel_launch(void* const* d_in, const int* in_sizes, int n_in,
                              void* d_out, int out_size, void* d_ws, size_t ws_size,
                              hipStream_t stream) {
  (void)in_sizes; (void)n_in; (void)out_size;
  const float* ue   = (const float*)d_in[0];
  const float* ie   = (const float*)d_in[1];
  const float* vals = (const float*)d_in[2];
  const int*   rows = (const int*)d_in[3];
  const int*   cols = (const int*)d_in[4];
  float* out = (float*)d_out;

  // layer 0 = concat(user_emb, item_emb), written in place in d_out
  k_layer0<<<(N_TOT * 32 + 255) / 256, 256, 0, stream>>>(ue, ie, out);

  const size_t ptr_bytes = (((size_t)(N_TOT + 1) * sizeof(int)) + 255) & ~(size_t)255;
  const size_t ws_needed = 2 * ptr_bytes + (size_t)NNZ_N * (sizeof(int) + sizeof(float));

  if (ws_size >= ws_needed) {
    // CSR path: no float atomics in the SpMM hot loop
    char* w = (char*)d_ws;
    int*   row_ptr  = (int*)w;   w += ptr_bytes;
    int*   row_fill = (int*)w;   w += ptr_bytes;
    int*   ccol     = (int*)w;   w += (size_t)NNZ_N * sizeof(int);
    float* cval     = (float*)w;

    k_zero<<<(N_TOT + 1 + 255) / 256, 256, 0, stream>>>(row_ptr, N_TOT + 1);
    k_hist<<<(NNZ_N + 255) / 256, 256, 0, stream>>>(rows, row_ptr);
    k_scan<<<1, 1024, 0, stream>>>(row_ptr, row_fill, N_TOT);
    k_scatter<<<(NNZ_N + 255) / 256, 256, 0, stream>>>(vals, rows, cols, row_fill, ccol, cval);
    for (int l = 1; l <= N_LAYERS; ++l)
      k_spmm<<<N_TOT / 8, 256, 0, stream>>>(row_ptr, ccol, cval, out, l);
  } else {
    // fallback: atomic COO scatter-add (no workspace needed)
    for (int l = 1; l <= N_LAYERS; ++l) {
      k_zero_layer<<<(N_TOT * 32 + 255) / 256, 256, 0, stream>>>(out, l);
      k_spmm_atomic<<<(NNZ_N * 32 + 255) / 256, 256, 0, stream>>>(vals, rows, cols, out, l);
    }
  }
}